// SGBM_54279796687063
// MI455X (gfx1250) — compile-verified
//
#include <hip/hip_runtime.h>
#include <hip/hip_bf16.h>

// ---------------- SGBM constants (match reference) ----------------
#define IMG_H 512
#define IMG_W 960
#define NDISP 96          // disparities 16..111
#define MIN_DISP 16
#define P1C 600.0f        // 8*3*5^2
#define P2C 2400.0f       // 32*3*5^2
#define BIGF 1.0e9f

typedef __attribute__((ext_vector_type(2))) float v2f;
typedef __attribute__((ext_vector_type(8))) float v8f;

// -------------------------------------------------------------------
// Kernel 1: masked absolute-difference cost (before box filter)
//   ad[y][x][d] = (x >= d+16) ? sum_c |L[y,x,c] - R[y,x-d-16,c]| : 765
// Layout: [H][W][D], D innermost. Block: 256 threads covers 32 x * 96 d
// of one image row; image tiles staged through LDS.
// -------------------------------------------------------------------
__global__ void sgbm_ad_kernel(const float* __restrict__ L,
                               const float* __restrict__ R,
                               float* __restrict__ out) {
  __shared__ float sL[32 * 3];
  __shared__ float sR[127 * 3];   // cols x0-111 .. x0+15

  const int x0 = blockIdx.x * 32;
  const int y  = blockIdx.y;
  const int tid = threadIdx.x;

  if (tid < 96) sL[tid] = L[(y * IMG_W + x0) * 3 + tid];
  for (int i = tid; i < 127 * 3; i += 256) {
    int xr = x0 - 111 + i / 3;
    int c  = i % 3;
    sR[i] = (xr >= 0 && xr < IMG_W) ? R[(y * IMG_W + xr) * 3 + c] : 0.0f;
  }
  __syncthreads();

  for (int i = tid; i < 32 * NDISP; i += 256) {
    int xi = i / NDISP;
    int d  = i - xi * NDISP;
    int x  = x0 + xi;
    float v;
    if (x < d + MIN_DISP) {
      v = 3.0f * 255.0f;
    } else {
      int rr = xi - d + 95;     // (x - (d+16)) - (x0 - 111)
      v = fabsf(sL[xi * 3 + 0] - sR[rr * 3 + 0])
        + fabsf(sL[xi * 3 + 1] - sR[rr * 3 + 1])
        + fabsf(sL[xi * 3 + 2] - sR[rr * 3 + 2]);
    }
    out[(y * IMG_W + x) * NDISP + d] = v;
  }
}

// -------------------------------------------------------------------
// Kernel 2/3: 5-tap band sum as WMMA band-GEMM.
//   Out[r][c] = sum_{k=0..4} In[r-2+k][c]   (zero outside row segment)
// Per 16x16 tile: Band[16x20] x In[20x16] via five V_WMMA_F32_16X16X4_F32.
// Segment-boundary zeros are folded into the *band coefficients* (they
// depend only on K), so B loads are unconditional from clamped rows:
// no exec-mask branching around loads. All indexing is 32-bit
// (47.2M elements < 2^31) so addressing is SADDR + 32-bit voffset.
//   horizontal pass: rows = y*W+x (stride NDISP, segment W)
//   vertical pass:   rows = y     (stride W*NDISP, segment H)
// -------------------------------------------------------------------
__global__ void sgbm_band5_wmma(const float* __restrict__ in,
                                float* __restrict__ out,
                                int colTiles, int rowStride, int seg) {
  const int wave = (int)((blockIdx.x * blockDim.x + threadIdx.x) >> 5);
  const int lane = threadIdx.x & 31;
  const int tR = wave / colTiles;
  const int tC = wave - tR * colTiles;

  const int r0 = tR * 16;
  const int c0 = tC * 16;
  const int segStart = (r0 / seg) * seg;
  const int segEnd   = segStart + seg;       // exclusive
  const int hi = lane >> 4;                  // 0 or 1
  const int n  = lane & 15;                  // tile column (N) == A row (M)
  const int col = c0 + n;

  v8f acc = {};
#pragma unroll
  for (int k0 = 0; k0 < 20; k0 += 4) {
    const int kx = k0 + hi * 2;      // A VGPR0 holds K=k0 (+2 for hi lanes)
    const int ky = kx + 1;           // A VGPR1 holds K=k0+1 (+2 for hi lanes)
    const int rx = r0 - 2 + kx;
    const int ry = r0 - 2 + ky;
    v2f a, b;
    // band matrix with boundary zeros folded in:
    // A[M][K] = 1 iff (M <= K <= M+4) and input row r0-2+K is in-segment
    a.x = (kx >= n && kx <= n + 4 && rx >= segStart && rx < segEnd) ? 1.0f : 0.0f;
    a.y = (ky >= n && ky <= n + 4 && ry >= segStart && ry < segEnd) ? 1.0f : 0.0f;
    // unconditional loads from clamped (always valid) rows
    const int rcx = min(max(rx, segStart), segEnd - 1);
    const int rcy = min(max(ry, segStart), segEnd - 1);
    b.x = in[rcx * rowStride + col];
    b.y = in[rcy * rowStride + col];
    acc = __builtin_amdgcn_wmma_f32_16x16x4_f32(
        /*neg_a=*/false, a, /*neg_b=*/false, b,
        /*c_mod=*/(short)0, acc, /*reuse_a=*/false, /*reuse_b=*/false);
  }
  // C/D layout: VGPR r -> M = r + 8*hi, N = lane&15
#pragma unroll
  for (int r = 0; r < 8; ++r) {
    out[(r0 + r + hi * 8) * rowStride + col] = acc[r];
  }
}

// -------------------------------------------------------------------
// Kernels 4-7: SGM DP scan along one direction.
//   cur[d] = c[d] + min(prev[d], m+P2, min(prev[d-1],prev[d+1])+P1) - m
// One 96-thread block (3 wave32) per scan line; prev vector in LDS,
// global min via wave32 shfl_xor reduction + 3-entry LDS combine.
// accumulate==0: agg = cur   accumulate==1: agg += cur
// -------------------------------------------------------------------
__device__ __forceinline__ float wave_min32(float v) {
#pragma unroll
  for (int i = 16; i > 0; i >>= 1) v = fminf(v, __shfl_xor(v, i, 32));
  return v;
}

__global__ void sgbm_scan_kernel(const float* __restrict__ cost,
                                 float* __restrict__ agg,
                                 int lineStride, int nSteps, int stepStride,
                                 int reverse, int accumulate) {
  __shared__ float sPrev[NDISP + 2];
  __shared__ float sWMin[3];

  const int d    = threadIdx.x;       // 0..95
  const int lane = d & 31;
  const int wid  = d >> 5;
  const int idx  = d + 1;
  const int base = blockIdx.x * lineStride;
  const int dir  = reverse ? -stepStride : stepStride;

  if (d == 0) { sPrev[0] = BIGF; sPrev[NDISP + 1] = BIGF; }

  // step 0: output is the raw cost
  const int pos0 = reverse ? (nSteps - 1) : 0;
  int o = base + pos0 * stepStride + d;
  float cur = cost[o];
  if (accumulate) agg[o] += cur; else agg[o] = cur;

  sPrev[idx] = cur;
  float wm = wave_min32(cur);
  if (lane == 0) sWMin[wid] = wm;
  __syncthreads();
  float m = fminf(fminf(sWMin[0], sWMin[1]), sWMin[2]);
  float p = cur;

  for (int s = 1; s < nSteps; ++s) {
    const float pm1 = sPrev[idx - 1];
    const float pp1 = sPrev[idx + 1];
    o += dir;
    const float c = cost[o];
    __builtin_prefetch(cost + o + dir, 0, 1);   // global_prefetch_b8 next step
    cur = c + fminf(fminf(p, m + P2C), fminf(pm1, pp1) + P1C) - m;
    if (accumulate) agg[o] += cur; else agg[o] = cur;
    __syncthreads();                 // neighbor reads done
    sPrev[idx] = cur;
    wm = wave_min32(cur);
    if (lane == 0) sWMin[wid] = wm;
    __syncthreads();                 // writes visible
    m = fminf(fminf(sWMin[0], sWMin[1]), sWMin[2]);
    p = cur;
  }
}

// -------------------------------------------------------------------
// Kernel 8: winner-take-all + parabolic sub-pixel refinement.
// One thread per pixel; 96 contiguous f32 read as float4 (384B aligned).
// -------------------------------------------------------------------
__global__ void sgbm_wta_kernel(const float* __restrict__ agg,
                                float* __restrict__ out) {
  const int p = blockIdx.x * blockDim.x + threadIdx.x;
  if (p >= IMG_H * IMG_W) return;
  const int base = p * NDISP;
  const float4* a4 = (const float4*)(agg + base);

  float best = BIGF;
  int bi = 0;
#pragma unroll 4
  for (int i = 0; i < NDISP / 4; ++i) {
    float4 v = a4[i];
    int b4 = i * 4;
    if (v.x < best) { best = v.x; bi = b4 + 0; }
    if (v.y < best) { best = v.y; bi = b4 + 1; }
    if (v.z < best) { best = v.z; bi = b4 + 2; }
    if (v.w < best) { best = v.w; bi = b4 + 3; }
  }
  const int im = bi > 0 ? bi - 1 : 0;
  const int ip = bi < NDISP - 1 ? bi + 1 : NDISP - 1;
  const float cm = agg[base + im];
  const float c0 = best;
  const float cp = agg[base + ip];
  const float denom = cm - 2.0f * c0 + cp;
  const float delta = (denom > 0.0f)
      ? (cm - cp) / (2.0f * fmaxf(denom, 1e-6f)) : 0.0f;
  out[p] = (float)MIN_DISP + (float)bi + delta;
}

// -------------------------------------------------------------------
// Host-side orchestration (all on `stream`, graph-capture safe).
// Workspace: two ping-pong volumes of H*W*D floats (189 MB each).
//   bufA: ad -> (hbox) bufB -> (vbox) bufA = final cost volume
//   bufB: agg accumulator (LR writes, RL/TB/BT add)
// -------------------------------------------------------------------
extern "C" void kernel_launch(void* const* d_in, const int* in_sizes, int n_in,
                              void* d_out, int out_size, void* d_ws, size_t ws_size,
                              hipStream_t stream) {
  const float* img_l = (const float*)d_in[0];
  const float* img_r = (const float*)d_in[1];
  float* out = (float*)d_out;

  const long VOL = (long)IMG_H * IMG_W * NDISP;   // 47,185,920 floats
  float* bufA = (float*)d_ws;
  float* bufB = bufA + VOL;

  // 1) masked SAD cost into bufA
  sgbm_ad_kernel<<<dim3(IMG_W / 32, IMG_H), 256, 0, stream>>>(img_l, img_r, bufA);

  // 2) horizontal 5-sum (rows = y*W+x, stride D, segment W): bufA -> bufB
  {
    const int rowTiles = (IMG_H * IMG_W) / 16;   // 30720
    const int colTiles = NDISP / 16;             // 6
    const int waves = rowTiles * colTiles;       // 184320 (exact /8)
    sgbm_band5_wmma<<<waves / 8, 256, 0, stream>>>(
        bufA, bufB, colTiles, NDISP, IMG_W);
  }

  // 3) vertical 5-sum (rows = y, stride W*D, segment H): bufB -> bufA
  {
    const int rowTiles = IMG_H / 16;             // 32
    const int colTiles = (IMG_W * NDISP) / 16;   // 5760
    const int waves = rowTiles * colTiles;       // 184320 (exact /8)
    sgbm_band5_wmma<<<waves / 8, 256, 0, stream>>>(
        bufB, bufA, colTiles, IMG_W * NDISP, IMG_H);
  }

  // 4-7) four-direction SGM aggregation into bufB
  const int WD = IMG_W * NDISP;
  // L -> R (initializes agg)
  sgbm_scan_kernel<<<IMG_H, NDISP, 0, stream>>>(bufA, bufB, WD, IMG_W,
                                                NDISP, 0, 0);
  // R -> L
  sgbm_scan_kernel<<<IMG_H, NDISP, 0, stream>>>(bufA, bufB, WD, IMG_W,
                                                NDISP, 1, 1);
  // T -> B
  sgbm_scan_kernel<<<IMG_W, NDISP, 0, stream>>>(bufA, bufB, NDISP, IMG_H,
                                                WD, 0, 1);
  // B -> T
  sgbm_scan_kernel<<<IMG_W, NDISP, 0, stream>>>(bufA, bufB, NDISP, IMG_H,
                                                WD, 1, 1);

  // 8) WTA + subpixel
  const int npix = IMG_H * IMG_W;
  sgbm_wta_kernel<<<(npix + 255) / 256, 256, 0, stream>>>(bufB, out);
}